// InterpLnr_41669772706019
// MI455X (gfx1250) — compile-verified
//
#include <hip/hip_runtime.h>

#ifndef __has_builtin
#define __has_builtin(x) 0
#endif

#define NSEG   7      // MAX_NUM_SEG
#define SEGL   64     // 2*MAX_LEN_SEG
#define TT     192    // T
#define CC     256    // C
#define OUTT   192    // MAX_LEN_PAD
#define XROWS  128    // rows of x staged in LDS (valid reads bounded by MAX_LEN_SEQ=128)
#define NTHR   256

#if defined(__AMDGCN__) && __has_builtin(__builtin_amdgcn_global_load_async_to_lds_b128)
#define HAVE_ASYNC_LDS 1
#else
#define HAVE_ASYNC_LDS 0
#endif

typedef int   v4i __attribute__((ext_vector_type(4)));
typedef float v4f __attribute__((ext_vector_type(4)));
typedef __attribute__((address_space(1))) v4i g_v4i;   // global int4
typedef __attribute__((address_space(3))) v4i l_v4i;   // LDS int4

// Dynamic LDS layout:
//   [0,768)        int   row_src[192]   source row index per output row (-1 = zero row)
//   [768,1536)     float row_lam[192]   lerp weight per output row
//   [1536,1568)    int   s_cnt[8]       per-segment valid counts -> exclusive prefix
//   [1568,1600)    int   s_off[8]       per-segment start offsets (cumsum of len_seg)
//   [2048,133120)  float xt[128*256]    staged x[b,0:128,:]
__global__ __launch_bounds__(NTHR) void InterpLnr_kernel(
    const float* __restrict__ x, const float* __restrict__ scales,
    const int* __restrict__ len_seg, const int* __restrict__ len_seq,
    float* __restrict__ out)
{
  extern __shared__ char smem[];
  int*   row_src = (int*)smem;
  float* row_lam = (float*)(smem + 768);
  int*   s_cnt   = (int*)(smem + 1536);
  int*   s_off   = (int*)(smem + 1568);
  float* xt      = (float*)(smem + 2048);

  const int b   = blockIdx.x;
  const int tid = threadIdx.x;

  // ---- Phase 0: kick off async stage of x[b, 0:128, :] (131072 B) into LDS ----
  const char* gsrc = (const char*)(x + (size_t)b * TT * CC);
  char*       ldst = (char*)xt;
  #pragma unroll
  for (int i = 0; i < (XROWS * CC * 4) / (16 * NTHR); ++i) {   // 32 x b128 per thread
    const int off = (tid + i * NTHR) * 16;
#if HAVE_ASYNC_LDS
    __builtin_amdgcn_global_load_async_to_lds_b128(
        (g_v4i*)(gsrc + off), (l_v4i*)(ldst + off), 0, 0);
#else
    *(v4f*)(ldst + off) = *(const v4f*)(gsrc + off);
#endif
  }

  // ---- Phase 1 (overlaps DMA): build the compaction mapping ----
  const float seqA = (float)(len_seq[b] - 1);
  if (tid < NSEG) {
    int offs = 0;
    for (int s = 0; s < tid; ++s) offs += len_seg[b * NSEG + s];
    s_off[tid] = offs;
    const float scale = scales[b * NSEG + tid];
    const float segA  = (float)(len_seg[b * NSEG + tid] - 1);
    const float offf  = (float)offs;
    int cnt = 0;
    for (int j = 0; j < SEGL; ++j) {
      float z  = (float)j / scale;
      float fl = floorf(z);
      if ((fl < segA) && (fl + offf < seqA)) cnt++;   // mask is a prefix in j
    }
    s_cnt[tid] = cnt;
  }
  if (tid < OUTT) row_src[tid] = -1;
  __syncthreads();
  if (tid == 0) {                                    // 7-wide exclusive scan
    int acc = 0;
    for (int s = 0; s < NSEG; ++s) { int c = s_cnt[s]; s_cnt[s] = acc; acc += c; }
  }
  __syncthreads();
  for (int k = tid; k < NSEG * SEGL; k += NTHR) {
    const int s = k >> 6, j = k & 63;
    const float scale = scales[b * NSEG + s];
    const float z   = (float)j / scale;
    const float fl  = floorf(z);
    const float lam = z - fl;
    const int   offs = s_off[s];
    const float segA = (float)(len_seg[b * NSEG + s] - 1);
    if ((fl < segA) && (fl + (float)offs < seqA)) {
      const int pos = s_cnt[s] + j;                  // prefix property => compacted slot
      if (pos < OUTT) {
        int idxB = (int)(fl + (float)offs);
        idxB = idxB < 0 ? 0 : (idxB > TT - 2 ? TT - 2 : idxB);
        row_src[pos] = idxB;
        row_lam[pos] = lam;
      }
    }
  }

  // ---- join: DMA complete (per-wave asynccnt) + mapping visible (barrier) ----
#if HAVE_ASYNC_LDS
#if __has_builtin(__builtin_amdgcn_s_wait_asynccnt)
  __builtin_amdgcn_s_wait_asynccnt(0);
#else
  asm volatile("s_wait_asynccnt 0" ::: "memory");
#endif
#endif
  __syncthreads();

  // ---- Phase 2: 64 lanes per output row; LDS gather + lerp + NT store ----
  float* outb = out + (size_t)b * OUTT * CC;
  const int lane = tid & 63;
  for (int p = tid >> 6; p < OUTT; p += NTHR / 64) {
    const int src = row_src[p];
    v4f r;
    if (src >= 0) {
      const float w  = row_lam[p];
      const float iw = 1.0f - w;
      const v4f a = *(const v4f*)(xt + src * CC + lane * 4);
      const v4f c = *(const v4f*)(xt + (src + 1) * CC + lane * 4);
      r = iw * a + w * c;
    } else {
      r = (v4f)(0.0f);
    }
    __builtin_nontemporal_store(r, (v4f*)(outb + (size_t)p * CC + lane * 4));
  }
}

extern "C" void kernel_launch(void* const* d_in, const int* in_sizes, int n_in,
                              void* d_out, int out_size, void* d_ws, size_t ws_size,
                              hipStream_t stream) {
  const float* x      = (const float*)d_in[0];
  const float* scales = (const float*)d_in[1];
  const int*   len_sg = (const int*)d_in[2];
  const int*   len_sq = (const int*)d_in[3];
  float* out = (float*)d_out;

  const int Bn = in_sizes[3];                        // 256 batch elements
  const size_t smem = 2048 + (size_t)XROWS * CC * 4; // 133120 B dynamic LDS
  (void)hipFuncSetAttribute((const void*)InterpLnr_kernel,
                            hipFuncAttributeMaxDynamicSharedMemorySize, (int)smem);
  InterpLnr_kernel<<<dim3(Bn), dim3(NTHR), smem, stream>>>(x, scales, len_sg, len_sq, out);
}